// SpatialAttention_1580547966423
// MI455X (gfx1250) — compile-verified
//
#include <hip/hip_runtime.h>

typedef __attribute__((ext_vector_type(2))) float v2f;
typedef __attribute__((ext_vector_type(8))) float v8f;

constexpr int Bn  = 32;
constexpr int IDF = 768;
constexpr int CDF = 768;
constexpr int Q   = 1024;   // IH*IW = 32*32
constexpr int SL  = 128;
constexpr size_t WC_ELEMS = (size_t)Bn * IDF * Q;   // weightedContext elems in d_out

__device__ __forceinline__ v8f wmma_f32(v2f a, v2f b, v8f c) {
  // V_WMMA_F32_16X16X4_F32 : D = A(16x4,f32) * B(4x16,f32) + C(16x16,f32)
  return __builtin_amdgcn_wmma_f32_16x16x4_f32(false, a, false, b, (short)0, c,
                                               false, false);
}

// -------- Kernel 1: sourceT[b] (768x128) = w_conv (768x768) @ context[b] (768x128)
// one wave computes a 64x32 tile; K = 768
__global__ __launch_bounds__(32) void k_source(const float* __restrict__ W,
                                               const float* __restrict__ ctx,
                                               float* __restrict__ srcT) {
  const int b    = blockIdx.z;
  const int m0   = blockIdx.x * 64;
  const int n0   = blockIdx.y * 32;
  const int lane = threadIdx.x;
  const int lm   = lane & 15;
  const int lk   = (lane >> 4) << 1;
  const int hs   = lane >> 4;
  const float* Bp = ctx + (size_t)b * CDF * SL;

  v8f acc[4][2] = {};
  for (int k0 = 0; k0 < CDF; k0 += 4) {
    v2f af[4];
#pragma unroll
    for (int i = 0; i < 4; ++i) {
      const float* ap = W + (size_t)(m0 + i * 16 + lm) * CDF + (k0 + lk);
      af[i][0] = ap[0]; af[i][1] = ap[1];
    }
    v2f bf[2];
#pragma unroll
    for (int j = 0; j < 2; ++j) {
      const float* bp = Bp + (size_t)(k0 + lk) * SL + (n0 + j * 16 + lm);
      bf[j][0] = bp[0]; bf[j][1] = bp[SL];
    }
#pragma unroll
    for (int i = 0; i < 4; ++i)
#pragma unroll
      for (int j = 0; j < 2; ++j)
        acc[i][j] = wmma_f32(af[i], bf[j], acc[i][j]);
  }

  float* cp = srcT + (size_t)b * IDF * SL;
#pragma unroll
  for (int i = 0; i < 4; ++i)
#pragma unroll
    for (int j = 0; j < 2; ++j)
#pragma unroll
      for (int r = 0; r < 8; ++r)
        cp[(size_t)(m0 + i * 16 + r + 8 * hs) * SL + (n0 + j * 16 + lm)] = acc[i][j][r];
}

// -------- Kernel 2: attn logits (16 q-rows x 128 s) per wave, fused mask+softmax,
// writes softmaxed attention TRANSPOSED ([b][s][q]) into attn_out region of d_out.
__global__ __launch_bounds__(32) void k_attn(const float* __restrict__ inp,
                                             const float* __restrict__ srcT,
                                             const int* __restrict__ mask,
                                             float* __restrict__ attnT) {
  const int b    = blockIdx.y;
  const int q0   = blockIdx.x * 16;
  const int lane = threadIdx.x;
  const int lm   = lane & 15;
  const int lk   = (lane >> 4) << 1;
  const int hs   = lane >> 4;
  const float* T = inp  + (size_t)b * IDF * Q;   // target[b]: (i, q), q contiguous
  const float* S = srcT + (size_t)b * IDF * SL;  // sourceT[b]: (i, s), s contiguous

  float negm[8];
#pragma unroll
  for (int t = 0; t < 8; ++t)
    negm[t] = -10000.0f * (float)mask[b * SL + t * 16 + lm];

  v8f acc[8] = {};
  for (int k0 = 0; k0 < IDF; k0 += 4) {
    // A(m=q, k=i) = target^T : element at i*Q + q (k-stride Q)
    v2f af;
    const float* ap = T + (size_t)(k0 + lk) * Q + (q0 + lm);
    af[0] = ap[0]; af[1] = ap[Q];
    v2f bf[8];
#pragma unroll
    for (int t = 0; t < 8; ++t) {
      const float* bp = S + (size_t)(k0 + lk) * SL + (t * 16 + lm);
      bf[t][0] = bp[0]; bf[t][1] = bp[SL];
    }
#pragma unroll
    for (int t = 0; t < 8; ++t) acc[t] = wmma_f32(af, bf[t], acc[t]);
  }

  float* aout = attnT + (size_t)b * SL * Q;
#pragma unroll
  for (int r = 0; r < 8; ++r) {
    const int q = q0 + r + 8 * hs;       // this lane-half's row for VGPR r
    float v[8];
    float mx = -3.4e38f;
#pragma unroll
    for (int t = 0; t < 8; ++t) { v[t] = acc[t][r] + negm[t]; mx = fmaxf(mx, v[t]); }
    // reduce across the 16-lane half (xor masks < 16 stay within the half)
#pragma unroll
    for (int off = 8; off >= 1; off >>= 1) mx = fmaxf(mx, __shfl_xor(mx, off));
    float sum = 0.0f;
#pragma unroll
    for (int t = 0; t < 8; ++t) { v[t] = __expf(v[t] - mx); sum += v[t]; }
#pragma unroll
    for (int off = 8; off >= 1; off >>= 1) sum += __shfl_xor(sum, off);
    const float inv = 1.0f / sum;
#pragma unroll
    for (int t = 0; t < 8; ++t)
      aout[(size_t)(t * 16 + lm) * Q + q] = v[t] * inv;   // store transposed [s][q]
  }
}

// -------- Kernel 3: weightedContext[b] (768x1024) = sourceT[b] (768x128) @ attnT[b] (128x1024)
// one wave computes a 64x64 tile; K = 128
__global__ __launch_bounds__(32) void k_wctx(const float* __restrict__ srcT,
                                             const float* __restrict__ attnT,
                                             float* __restrict__ out) {
  const int b    = blockIdx.z;
  const int m0   = blockIdx.x * 64;
  const int n0   = blockIdx.y * 64;
  const int lane = threadIdx.x;
  const int lm   = lane & 15;
  const int lk   = (lane >> 4) << 1;
  const int hs   = lane >> 4;
  const float* S = srcT  + (size_t)b * IDF * SL;
  const float* P = attnT + (size_t)b * SL * Q;

  v8f acc[4][4] = {};
  for (int k0 = 0; k0 < SL; k0 += 4) {
    v2f af[4];
#pragma unroll
    for (int i = 0; i < 4; ++i) {
      const float* ap = S + (size_t)(m0 + i * 16 + lm) * SL + (k0 + lk);
      af[i][0] = ap[0]; af[i][1] = ap[1];
    }
    v2f bf[4];
#pragma unroll
    for (int j = 0; j < 4; ++j) {
      const float* bp = P + (size_t)(k0 + lk) * Q + (n0 + j * 16 + lm);
      bf[j][0] = bp[0]; bf[j][1] = bp[Q];
    }
#pragma unroll
    for (int i = 0; i < 4; ++i)
#pragma unroll
      for (int j = 0; j < 4; ++j)
        acc[i][j] = wmma_f32(af[i], bf[j], acc[i][j]);
  }

  float* C = out + (size_t)b * IDF * Q;
#pragma unroll
  for (int i = 0; i < 4; ++i)
#pragma unroll
    for (int j = 0; j < 4; ++j)
#pragma unroll
      for (int r = 0; r < 8; ++r)
        C[(size_t)(m0 + i * 16 + r + 8 * hs) * Q + (n0 + j * 16 + lm)] = acc[i][j][r];
}

extern "C" void kernel_launch(void* const* d_in, const int* in_sizes, int n_in,
                              void* d_out, int out_size, void* d_ws, size_t ws_size,
                              hipStream_t stream) {
  const float* inp  = (const float*)d_in[0];   // (B, IDF, IH, IW) fp32
  const float* ctx  = (const float*)d_in[1];   // (B, CDF, SL)     fp32
  const int*   mask = (const int*)d_in[2];     // (B, SL)          int32
  const float* wcv  = (const float*)d_in[3];   // (IDF, CDF)       fp32

  float* out   = (float*)d_out;
  float* srcT  = (float*)d_ws;          // B*IDF*SL fp32 = 12.6 MB scratch
  float* attnT = out + WC_ELEMS;        // attn_out region, [b][s][q] — also GEMM3's B

  k_source<<<dim3(IDF / 64, SL / 32, Bn), 32, 0, stream>>>(wcv, ctx, srcT);
  k_attn  <<<dim3(Q / 16, Bn),            32, 0, stream>>>(inp, srcT, mask, attnT);
  k_wctx  <<<dim3(IDF / 64, Q / 64, Bn),  32, 0, stream>>>(srcT, attnT, out);
}